// StratifiedSumPooling_73048803770493
// MI455X (gfx1250) — compile-verified
//
#include <hip/hip_runtime.h>

// CDNA5 / gfx1250 — StratifiedSumPooling as one-hot GEMM via V_WMMA_F32_16X16X4_F32.
//
// Out[8192 x 100] = V[8192 x 4096] x OneHot(labels)[4096 x 100]
// fp32 WMMA keeps the math exact vs the fp32 reference (B entries are 0.0/1.0).
// Each wave carries TWO 16-row tiles so the one-hot B fragment (14 cmps +
// 14 cndmasks per k-slab) is amortized over 14 WMMAs instead of 7.

typedef float v2f __attribute__((ext_vector_type(2)));
typedef float v8f __attribute__((ext_vector_type(8)));

#define BATCH       8192
#define PROTOS      4096
#define CLASSES     100
#define NTILES      7                  // 7 * 16 = 112 >= 100 classes
#define WAVES       8                  // waves per block (wave32)
#define THREADS     (WAVES * 32)
#define KSLABS      (PROTOS / 4)       // 1024 slabs of K=4
#define SLABS_PER_W (KSLABS / WAVES)   // 128 slabs per wave (K split 8-ways)
#define ROWTILES    2                  // 16-row tiles per wave
#define BLOCK_ROWS  (16 * ROWTILES)    // 32 rows per block

__global__ __launch_bounds__(THREADS)
void stratified_sum_wmma(const float* __restrict__ V,
                         const int*   __restrict__ labels,
                         float*       __restrict__ out) {
    // Phase 1: first 4096 ints = LDS label cache (16 KB).
    // Phase 2: 8 waves * 7 tiles * 32 lanes * 8 regs floats = 56 KB partials,
    //          reused once per row tile. Aliasing guarded by barriers.
    __shared__ __align__(32) float smem[WAVES * NTILES * 32 * 8];
    int* lds_labels = (int*)smem;

    const int tid  = threadIdx.x;
    const int wid  = tid >> 5;
    const int lane = tid & 31;
    const int row_base = blockIdx.x * BLOCK_ROWS;

    // ---- Phase 0: coalesced copy of labels into LDS ----
    for (int i = tid; i < PROTOS; i += THREADS)
        lds_labels[i] = labels[i];
    __syncthreads();

    // ISA layout for 16x16x4 f32:
    //   A (16x4): lanes 0-15 = rows, v0=K0 v1=K1 ; lanes 16-31 = rows, v0=K2 v1=K3
    //   B (4x16): lanes 0-15 = cols, v0=K0 v1=K1 ; lanes 16-31 = cols, v0=K2 v1=K3
    const int l15  = lane & 15;
    const int koff = (lane >> 4) * 2;          // 0 or 2

    v8f acc[ROWTILES][NTILES];
#pragma unroll
    for (int h = 0; h < ROWTILES; ++h)
#pragma unroll
        for (int t = 0; t < NTILES; ++t) { v8f z = {}; acc[h][t] = z; }

    const int    kstart = wid * SLABS_PER_W * 4;
    const float* vp0 = V + (size_t)(row_base + l15) * PROTOS + kstart + koff;
    const float* vp1 = vp0 + (size_t)16 * PROTOS;   // second row tile
    const int*   lp  = lds_labels + kstart + koff;

    // ---- Main K loop: 128 slabs, 14 WMMAs each (B shared across row tiles) ----
#pragma unroll 4
    for (int s = 0; s < SLABS_PER_W; ++s) {
        v2f a0 = *(const v2f*)vp0;             // global_load_b64 per row tile
        v2f a1 = *(const v2f*)vp1;
        const int la = lp[0];                  // ds_load_b64
        const int lb = lp[1];
#pragma unroll
        for (int t = 0; t < NTILES; ++t) {
            const int cls = t * 16 + l15;
            v2f b;
            b.x = (la == cls) ? 1.0f : 0.0f;   // one-hot B fragment (2 VALU / element)
            b.y = (lb == cls) ? 1.0f : 0.0f;
            acc[0][t] = __builtin_amdgcn_wmma_f32_16x16x4_f32(
                false, a0, false, b, (short)0, acc[0][t], false, false);
            acc[1][t] = __builtin_amdgcn_wmma_f32_16x16x4_f32(
                false, a1, false, b, (short)0, acc[1][t], false, false);
        }
        vp0 += 4;
        vp1 += 4;
        lp  += 4;
    }

    // ---- Phases 2/3 per row tile: fixed-order 8-way K reduction + store ----
    // D layout: acc[.][t] reg r: lanes 0-15 -> row r,   class 16t+lane
    //                            lanes 16-31 -> row r+8, class 16t+lane-16
#pragma unroll
    for (int h = 0; h < ROWTILES; ++h) {
        __syncthreads();   // h==0: label reads done; h==1: previous reduction done
#pragma unroll
        for (int t = 0; t < NTILES; ++t) {
            float* dst = &smem[(((wid * NTILES) + t) * 32 + lane) * 8];
            *(v8f*)dst = acc[h][t];            // 2x ds_store_b128
        }
        __syncthreads();

        for (int idx = tid; idx < NTILES * 256; idx += THREADS) {
            const int t  = idx >> 8;
            const int ln = (idx >> 3) & 31;
            const int r  = idx & 7;
            float sum = 0.0f;
#pragma unroll
            for (int w = 0; w < WAVES; ++w)
                sum += smem[(((w * NTILES) + t) * 32 + ln) * 8 + r];
            const int row = row_base + h * 16 + r + ((ln >> 4) << 3);
            const int cls = t * 16 + (ln & 15);
            if (cls < CLASSES)
                out[(size_t)row * CLASSES + cls] = sum;
        }
    }
}

extern "C" void kernel_launch(void* const* d_in, const int* in_sizes, int n_in,
                              void* d_out, int out_size, void* d_ws, size_t ws_size,
                              hipStream_t stream) {
    const float* V      = (const float*)d_in[0];   // values [8192, 4096] fp32
    const int*   labels = (const int*)d_in[1];     // labels [4096] int
    float*       out    = (float*)d_out;           // out    [8192, 100] fp32

    stratified_sum_wmma<<<BATCH / BLOCK_ROWS, THREADS, 0, stream>>>(V, labels, out);
}